// MambaBlock_Temporal_31353261261555
// MI455X (gfx1250) — compile-verified
//
#include <hip/hip_runtime.h>
#include <cmath>

// ---------------- problem constants ----------------
#define NB    8      // batch
#define HID_  512
#define DIN_  1024
#define DST_  128    // DSTATE
#define DTR_  32     // DTRANK
#define LSEQ  192
#define XPN   288    // DTRANK + 2*DSTATE

typedef __attribute__((ext_vector_type(2))) float v2f;
typedef __attribute__((ext_vector_type(8))) float v8f;

__device__ __forceinline__ float siluf(float x)  { return x / (1.0f + __expf(-x)); }
__device__ __forceinline__ float softplusf(float x) { return (x > 20.0f) ? x : log1pf(__expf(x)); }

// ============================================================
// Generic batched GEMM via V_WMMA_F32_16X16X4_F32 (wave32).
// C[M,N] = act(A[M,K](row-major,lda) * W[N,K]^T(row-major,ldw) + bias[N])
// One wave computes one 16x16 tile.
// A 16x4 f32 layout (ISA 7.12.2): lane -> row m=lane%16, holds K = 2*(lane/16)+{0,1}
// B 4x16 mirrors with N across lanes. C/D: VGPR r -> row r + 8*(lane/16), col lane%16.
// ACT: 0 = none, 1 = softplus
// ============================================================
template<int ACT>
__global__ __launch_bounds__(256) void gemm_wmma_f32(
    const float* __restrict__ A, int lda, long strideA,
    const float* __restrict__ W, int ldw, long strideW,
    const float* __restrict__ bias,
    float* __restrict__ C, int ldc, long strideC,
    int M, int N, int K)
{
  const int batch = blockIdx.y;
  A += (long)batch * strideA;
  W += (long)batch * strideW;
  C += (long)batch * strideC;

  const int wave = blockIdx.x * (blockDim.x >> 5) + (threadIdx.x >> 5);
  const int lane = threadIdx.x & 31;
  const int tilesN = N >> 4;
  const int tm = wave / tilesN;
  const int tn = wave - tm * tilesN;
  if (tm * 16 >= M) return;                // wave-uniform: EXEC stays all-ones

  const int r16   = lane & 15;
  const int khalf = (lane >> 4) << 1;      // 0 or 2

  const float* Arow = A + (long)(tm * 16 + r16) * lda + khalf;
  const float* Wrow = W + (long)(tn * 16 + r16) * ldw + khalf;

  v8f acc = {};
  for (int k = 0; k < K; k += 4) {
    v2f av = *(const v2f*)(Arow + k);
    v2f bv = *(const v2f*)(Wrow + k);
    acc = __builtin_amdgcn_wmma_f32_16x16x4_f32(
        false, av, false, bv, (short)0, acc, false, false);
  }

  const int col     = tn * 16 + r16;
  const int rowBase = tm * 16 + ((lane >> 4) << 3);
  const float bcol  = bias ? bias[col] : 0.0f;
#pragma unroll
  for (int r = 0; r < 8; ++r) {
    float v = acc[r] + bcol;
    if (ACT == 1) v = softplusf(v);
    C[(long)(rowBase + r) * ldc + col] = v;
  }
}

// ============================================================
// gap8: 8x8 adaptive average pool of [*,64,64] planes.
// One 64-thread block per (img,b,c) plane; thread t owns output cell t
// (deterministic fixed-order summation, no atomics).
// pooled layout: [img][b][c][64]
// ============================================================
__global__ __launch_bounds__(64) void gap8_kernel(
    const float* __restrict__ x1, const float* __restrict__ x2,
    const float* __restrict__ x3, float* __restrict__ pooled)
{
  const int plane = blockIdx.x;                 // img*(NB*HID_) + b*HID_ + c
  const int img   = plane / (NB * HID_);
  const int rem   = plane - img * (NB * HID_);
  const float* src = (img == 0 ? x1 : (img == 1 ? x2 : x3)) + (long)rem * 4096;

  const int t  = threadIdx.x;                   // 0..63 = output cell
  const int oy = t >> 3, ox = t & 7;
  float s = 0.0f;
#pragma unroll
  for (int r = 0; r < 8; ++r) {
    const float* row = src + (oy * 8 + r) * 64 + ox * 8;
#pragma unroll
    for (int c = 0; c < 8; ++c) s += row[c];
  }
  pooled[(long)plane * 64 + t] = s * (1.0f / 64.0f);
}

// ============================================================
// Depthwise causal conv1d (k=4, left pad 3) + bias + SiLU.
// xz: [B*L, 2048] (xc = first 1024 cols) -> xconv: [B*L, 1024]
// ============================================================
__global__ __launch_bounds__(256) void conv1d_silu_kernel(
    const float* __restrict__ xz, const float* __restrict__ w,
    const float* __restrict__ bias, float* __restrict__ out)
{
  const int idx = blockIdx.x * 256 + threadIdx.x;   // (b*L + l)*DIN + c
  const int c = idx & (DIN_ - 1);
  const int l = (idx >> 10) % LSEQ;
  const int b = idx / (DIN_ * LSEQ);

  float s = bias[c];
  const float* wc = w + c * 4;
#pragma unroll
  for (int j = 0; j < 4; ++j) {
    const int ls = l - 3 + j;
    if (ls >= 0) s += wc[j] * xz[((long)(b * LSEQ + ls) << 11) + c];
  }
  out[idx] = siluf(s);
}

// ============================================================
// Selective scan. One wave per (b,d); 4 state elems per lane (DSTATE=128).
// Fused: y = (scan_out + xconv*D) * silu(z); writes ybuf [B*L, 1024].
// ============================================================
__global__ __launch_bounds__(256) void scan_kernel(
    const float* __restrict__ xz,    const float* __restrict__ xconv,
    const float* __restrict__ xdbl,  const float* __restrict__ delta,
    const float* __restrict__ A_log, const float* __restrict__ Dp,
    float* __restrict__ ybuf)
{
  const int wid  = blockIdx.x * (blockDim.x >> 5) + (threadIdx.x >> 5);
  const int lane = threadIdx.x & 31;
  const int b = wid >> 10;
  const int d = wid & (DIN_ - 1);
  if (b >= NB) return;

  const int n0 = lane * 4;
  float a0 = -__expf(A_log[(long)d * DST_ + n0 + 0]);
  float a1 = -__expf(A_log[(long)d * DST_ + n0 + 1]);
  float a2 = -__expf(A_log[(long)d * DST_ + n0 + 2]);
  float a3 = -__expf(A_log[(long)d * DST_ + n0 + 3]);
  const float dcoef = Dp[d];

  float h0 = 0.f, h1 = 0.f, h2 = 0.f, h3 = 0.f;
  for (int l = 0; l < LSEQ; ++l) {
    const long row = (long)b * LSEQ + l;
    const float dt = delta[row * DIN_ + d];
    const float xv = xconv[row * DIN_ + d];
    const float* xd = xdbl + row * XPN;
    const float4 Bv = *(const float4*)(xd + DTR_ + n0);          // B_t
    const float4 Cv = *(const float4*)(xd + DTR_ + DST_ + n0);   // C_t
    const float dx = dt * xv;
    h0 = h0 * __expf(dt * a0) + dx * Bv.x;
    h1 = h1 * __expf(dt * a1) + dx * Bv.y;
    h2 = h2 * __expf(dt * a2) + dx * Bv.z;
    h3 = h3 * __expf(dt * a3) + dx * Bv.w;
    float p = h0 * Cv.x + h1 * Cv.y + h2 * Cv.z + h3 * Cv.w;
#pragma unroll
    for (int off = 16; off; off >>= 1) p += __shfl_xor(p, off, 32);
    if (lane == 0) {
      float y = p + xv * dcoef;
      const float z = xz[(row << 11) + DIN_ + d];
      y *= siluf(z);
      ybuf[row * DIN_ + d] = y;
    }
  }
}

// ============================================================
// Transpose [B,L,512] -> [B,512,L] fused with MaxPool1d(3): out [B,512,64]
// ============================================================
__global__ __launch_bounds__(256) void maxpool_kernel(
    const float* __restrict__ yout, float* __restrict__ maxp)
{
  const int idx = blockIdx.x * 256 + threadIdx.x;  // b*512*64 + c*64 + t
  const int t = idx & 63;
  const int c = (idx >> 6) & (HID_ - 1);
  const int b = idx >> 15;
  float m = -INFINITY;
#pragma unroll
  for (int j = 0; j < 3; ++j)
    m = fmaxf(m, yout[((long)b * LSEQ + 3 * t + j) * HID_ + c]);
  maxp[idx] = m;
}

// ============================================================
// 3x3 conv, 512->512 channels, 8x8 spatial, pad 1, + bias.
// Grid (co_tile=8, b=8); block 256. Input ci-chunks of 128 cached in LDS.
// Thread: fixed position p = t&63, co = coBase + (t>>6) + 4*k, k=0..15.
// ============================================================
__global__ __launch_bounds__(256) void conv3x3_kernel(
    const float* __restrict__ in, const float* __restrict__ w,
    const float* __restrict__ bias, float* __restrict__ out)
{
  __shared__ float sin_[128 * 64];
  const int b      = blockIdx.y;
  const int coBase = blockIdx.x * 64;
  const int t = threadIdx.x;
  const int p = t & 63, y = p >> 3, x = p & 7;
  const int coOff = t >> 6;                         // 0..3

  float acc[16];
#pragma unroll
  for (int k = 0; k < 16; ++k) acc[k] = 0.0f;

  for (int cc = 0; cc < HID_; cc += 128) {
    __syncthreads();
    for (int i = t; i < 128 * 64; i += 256)
      sin_[i] = in[((long)b * HID_ + cc + (i >> 6)) * 64 + (i & 63)];
    __syncthreads();

    for (int ci = 0; ci < 128; ++ci) {
      float v[9];
      int vi = 0;
#pragma unroll
      for (int dy = -1; dy <= 1; ++dy)
#pragma unroll
        for (int dx = -1; dx <= 1; ++dx) {
          const int yy = y + dy, xx = x + dx;
          v[vi++] = (yy >= 0 && yy < 8 && xx >= 0 && xx < 8)
                      ? sin_[ci * 64 + yy * 8 + xx] : 0.0f;
        }
#pragma unroll
      for (int k = 0; k < 16; ++k) {
        const float* wp = w + ((long)(coBase + coOff + 4 * k) * HID_ + cc + ci) * 9;
        acc[k] += v[0]*wp[0] + v[1]*wp[1] + v[2]*wp[2]
                + v[3]*wp[3] + v[4]*wp[4] + v[5]*wp[5]
                + v[6]*wp[6] + v[7]*wp[7] + v[8]*wp[8];
      }
    }
  }
#pragma unroll
  for (int k = 0; k < 16; ++k) {
    const int co = coBase + coOff + 4 * k;
    out[((long)b * HID_ + co) * 64 + p] = acc[k] + bias[co];
  }
}

// ============================================================
// Bilinear upsample x8, align_corners=True: [B,512,8,8] -> [B,512,64,64]
// ============================================================
__global__ __launch_bounds__(256) void upsample_kernel(
    const float* __restrict__ in, float* __restrict__ out)
{
  const long idx = (long)blockIdx.x * 256 + threadIdx.x;   // 2^24 total
  const int ox = (int)(idx & 63);
  const int oy = (int)((idx >> 6) & 63);
  const long bc = idx >> 12;                               // b*512 + c

  const float s = 7.0f / 63.0f;
  const float cy = oy * s, cx = ox * s;
  int i0 = (int)floorf(cy); int i1 = i0 + 1 < 7 ? i0 + 1 : 7;
  int j0 = (int)floorf(cx); int j1 = j0 + 1 < 7 ? j0 + 1 : 7;
  const float wy = cy - (float)i0, wx = cx - (float)j0;

  const float* pl = in + bc * 64;
  const float v0 = pl[i0 * 8 + j0] * (1.0f - wx) + pl[i0 * 8 + j1] * wx;
  const float v1 = pl[i1 * 8 + j0] * (1.0f - wx) + pl[i1 * 8 + j1] * wx;
  out[idx] = v0 * (1.0f - wy) + v1 * wy;
}

// ============================================================
extern "C" void kernel_launch(void* const* d_in, const int* in_sizes, int n_in,
                              void* d_out, int out_size, void* d_ws, size_t ws_size,
                              hipStream_t stream) {
  (void)in_sizes; (void)n_in; (void)out_size; (void)ws_size;

  const float* x1        = (const float*)d_in[0];
  const float* x2        = (const float*)d_in[1];
  const float* x3        = (const float*)d_in[2];
  const float* a_[3]     = { (const float*)d_in[3], (const float*)d_in[4], (const float*)d_in[5] };
  const float* in_proj_w = (const float*)d_in[6];
  const float* conv1d_w  = (const float*)d_in[7];
  const float* conv1d_b  = (const float*)d_in[8];
  const float* x_proj_w  = (const float*)d_in[9];
  const float* dt_w      = (const float*)d_in[10];
  const float* dt_b      = (const float*)d_in[11];
  const float* A_log     = (const float*)d_in[12];
  const float* Dp        = (const float*)d_in[13];
  const float* out_proj_w= (const float*)d_in[14];
  const float* conv_w    = (const float*)d_in[15];
  const float* conv_b    = (const float*)d_in[16];
  float* outp            = (float*)d_out;

  // workspace layout (floats)
  float* ws     = (float*)d_ws;
  float* pooled = ws;                       // 3*8*512*64      =  786432
  float* Xs     = pooled + 786432;          // 8*192*512       =  786432
  float* XZ     = Xs + 786432;              // 1536*2048       = 3145728
  float* xconv  = XZ + 3145728;             // 1536*1024       = 1572864
  float* xdbl   = xconv + 1572864;          // 1536*288        =  442368
  float* delta  = xdbl + 442368;            // 1536*1024       = 1572864
  float* ybuf   = delta + 1572864;          // 1536*1024       = 1572864
  float* yout   = ybuf + 1572864;           // 1536*512        =  786432
  float* maxp   = yout + 786432;            // 8*512*64        =  262144
  float* cvout  = maxp + 262144;            // 8*512*64        =  262144

  const int MROWS = NB * LSEQ;              // 1536

  // 1) 8x8 average pool of x1..x3 -> pooled [3][B][512][64]
  gap8_kernel<<<dim3(3 * NB * HID_), dim3(64), 0, stream>>>(x1, x2, x3, pooled);

  // 2) tokens: t[b,l,c] = sum_k a[b,l,k]*pooled[b,c,k]  (WMMA, batched over b)
  for (int i = 0; i < 3; ++i) {
    gemm_wmma_f32<0><<<dim3(16, NB), 256, 0, stream>>>(
        a_[i], 64, (long)64 * 64,
        pooled + (long)i * NB * HID_ * 64, 64, (long)HID_ * 64,
        nullptr,
        Xs + (long)i * 64 * HID_, HID_, (long)LSEQ * HID_,
        64, HID_, 64);
  }

  // 3) in_proj: XZ[1536,2048] = Xs[1536,512] * in_proj_w[2048,512]^T
  gemm_wmma_f32<0><<<dim3((MROWS / 16) * (2 * DIN_ / 16) / 8, 1), 256, 0, stream>>>(
      Xs, HID_, 0, in_proj_w, HID_, 0, nullptr, XZ, 2 * DIN_, 0,
      MROWS, 2 * DIN_, HID_);

  // 4) depthwise causal conv1d + bias + SiLU -> xconv [1536,1024]
  conv1d_silu_kernel<<<dim3(MROWS * DIN_ / 256), 256, 0, stream>>>(
      XZ, conv1d_w, conv1d_b, xconv);

  // 5) x_proj: xdbl[1536,288] = xconv * x_proj_w[288,1024]^T
  gemm_wmma_f32<0><<<dim3((MROWS / 16) * (XPN / 16) / 8, 1), 256, 0, stream>>>(
      xconv, DIN_, 0, x_proj_w, DIN_, 0, nullptr, xdbl, XPN, 0,
      MROWS, XPN, DIN_);

  // 6) dt_proj + softplus: delta[1536,1024] = softplus(xdbl[:, :32]*dt_w^T + dt_b)
  gemm_wmma_f32<1><<<dim3((MROWS / 16) * (DIN_ / 16) / 8, 1), 256, 0, stream>>>(
      xdbl, XPN, 0, dt_w, DTR_, 0, dt_b, delta, DIN_, 0,
      MROWS, DIN_, DTR_);

  // 7) selective scan + D-skip + SiLU(z) gating -> ybuf [1536,1024]
  scan_kernel<<<dim3(NB * DIN_ / 8), 256, 0, stream>>>(
      XZ, xconv, xdbl, delta, A_log, Dp, ybuf);

  // 8) out_proj: yout[1536,512] = ybuf * out_proj_w[512,1024]^T
  gemm_wmma_f32<0><<<dim3((MROWS / 16) * (HID_ / 16) / 8, 1), 256, 0, stream>>>(
      ybuf, DIN_, 0, out_proj_w, DIN_, 0, nullptr, yout, HID_, 0,
      MROWS, HID_, DIN_);

  // 9) transpose + MaxPool1d(3) -> maxp [B,512,8,8]
  maxpool_kernel<<<dim3(NB * HID_ * 64 / 256), 256, 0, stream>>>(yout, maxp);

  // 10) 3x3 conv 512->512 + bias -> cvout [B,512,8,8]
  conv3x3_kernel<<<dim3(HID_ / 64, NB), 256, 0, stream>>>(maxp, conv_w, conv_b, cvout);

  // 11) bilinear x8 upsample (align_corners) -> d_out [B,512,64,64]
  upsample_kernel<<<dim3(NB * HID_ * 64 * 64 / 256), 256, 0, stream>>>(cvout, outp);
}